// GATClassifier_60284160967393
// MI455X (gfx1250) — compile-verified
//
#include <hip/hip_runtime.h>
#include <hip/hip_bf16.h>
#include <math.h>

typedef float v2f __attribute__((ext_vector_type(2)));
typedef float v8f __attribute__((ext_vector_type(8)));

#define HEADS 4
#define HID 32
#define F 128
#define NUM_GRAPHS 64
#define OUT_CH 10
#define NEG_SLOPE 0.2f

// ---- helpers ---------------------------------------------------------------

// order-preserving float<->uint encoding so atomicMax(u32) == float max
__device__ __forceinline__ unsigned fenc(float f) {
    unsigned u = __float_as_uint(f);
    return (u & 0x80000000u) ? ~u : (u | 0x80000000u);
}
__device__ __forceinline__ float fdec(unsigned u) {
    return __uint_as_float((u & 0x80000000u) ? (u & 0x7FFFFFFFu) : ~u);
}
__device__ __forceinline__ float lrelu(float v) { return v > 0.f ? v : NEG_SLOPE * v; }

// ---- GEMM: H[N,128] = X[N,128] @ W[128,128] via V_WMMA_F32_16X16X4_F32 -----
// block = 256 threads = 8 waves; block b covers rows [16b,16b+16),
// wave w covers cols [16w, 16w+16). N is a multiple of 16 (50000 = 3125*16).
__global__ __launch_bounds__(256) void gemm_xw_wmma(
    const float* __restrict__ X, const float* __restrict__ W,
    float* __restrict__ H) {
    const int wave = threadIdx.x >> 5;
    const int lane = threadIdx.x & 31;
    const int row0 = blockIdx.x * 16;
    const int col0 = wave * 16;
    const int m    = lane & 15;          // A: row index / B: col index
    const int kb   = (lane >> 4) << 1;   // 0 for lanes 0-15, 2 for lanes 16-31

    const float* xrow = X + (size_t)(row0 + m) * F;
    v8f acc = {};
    for (int k0 = 0; k0 < F; k0 += 4) {
        v2f a, b;
        a.x = xrow[k0 + kb];
        a.y = xrow[k0 + kb + 1];
        b.x = W[(size_t)(k0 + kb) * F + col0 + m];
        b.y = W[(size_t)(k0 + kb + 1) * F + col0 + m];
        acc = __builtin_amdgcn_wmma_f32_16x16x4_f32(
            /*neg_a=*/false, a, /*neg_b=*/false, b,
            /*c_mod=*/(short)0, acc, /*reuse_a=*/false, /*reuse_b=*/false);
    }
    // C/D layout: VGPR r -> M = r + (lane<16 ? 0 : 8), N = lane&15
    const int mrow = (lane >> 4) * 8;
    float* hout = H + (size_t)(row0 + mrow) * F + col0 + (lane & 15);
#pragma unroll
    for (int r = 0; r < 8; ++r) hout[(size_t)r * F] = acc[r];
}

// ---- per-node attention coefficients --------------------------------------
__global__ __launch_bounds__(256) void attn_coef(
    const float* __restrict__ H, const float* __restrict__ a_s,
    const float* __restrict__ a_d, float* __restrict__ asrc,
    float* __restrict__ adst, int n) {
    int idx = blockIdx.x * blockDim.x + threadIdx.x;   // node*4 + head
    if (idx >= n * HEADS) return;
    const int node = idx >> 2, head = idx & 3;
    const float* hp = H + (size_t)node * F + head * HID;
    const float* as = a_s + head * HID;
    const float* ad = a_d + head * HID;
    float s = 0.f, d = 0.f;
#pragma unroll
    for (int c = 0; c < HID; ++c) { float v = hp[c]; s += v * as[c]; d += v * ad[c]; }
    asrc[idx] = s;
    adst[idx] = d;
}

// ---- reset accumulators ----------------------------------------------------
__global__ __launch_bounds__(256) void reset_layer(
    unsigned* __restrict__ maxu, float* __restrict__ denom,
    float* __restrict__ agg, int n) {
    int idx = blockIdx.x * blockDim.x + threadIdx.x;
    if (idx < n * F) agg[idx] = 0.f;
    if (idx < n * HEADS) { maxu[idx] = 0x007FFFFFu /* fenc(-inf) */; denom[idx] = 0.f; }
}

// ---- edge passes -----------------------------------------------------------
__device__ __forceinline__ void edge_sd(int e, int E, const int* __restrict__ src,
                                        const int* __restrict__ dst, int& s, int& d) {
    if (e < E) { s = src[e]; d = dst[e]; }
    else       { s = d = e - E; }   // self loops appended after edges
}

__global__ __launch_bounds__(256) void edge_max_pass(
    const int* __restrict__ src, const int* __restrict__ dst,
    const float* __restrict__ asrc, const float* __restrict__ adst,
    unsigned* __restrict__ maxu, int E, int EN) {
    int e = blockIdx.x * blockDim.x + threadIdx.x;
    if (e >= EN) return;
    int s, d; edge_sd(e, E, src, dst, s, d);
#pragma unroll
    for (int h = 0; h < HEADS; ++h) {
        float v = lrelu(asrc[s * HEADS + h] + adst[d * HEADS + h]);
        atomicMax(&maxu[d * HEADS + h], fenc(v));
    }
}

__global__ __launch_bounds__(256) void edge_sum_pass(
    const int* __restrict__ src, const int* __restrict__ dst,
    const float* __restrict__ asrc, const float* __restrict__ adst,
    const unsigned* __restrict__ maxu, float* __restrict__ denom, int E, int EN) {
    int e = blockIdx.x * blockDim.x + threadIdx.x;
    if (e >= EN) return;
    int s, d; edge_sd(e, E, src, dst, s, d);
#pragma unroll
    for (int h = 0; h < HEADS; ++h) {
        float v = lrelu(asrc[s * HEADS + h] + adst[d * HEADS + h]);
        float m = fdec(maxu[d * HEADS + h]);
        atomicAdd(&denom[d * HEADS + h], __expf(v - m));
    }
}

// one wave (32 lanes) per edge; lane l handles channels 4l..4l+3 (head = l>>3)
__global__ __launch_bounds__(256) void edge_agg_pass(
    const int* __restrict__ src, const int* __restrict__ dst,
    const float* __restrict__ asrc, const float* __restrict__ adst,
    const unsigned* __restrict__ maxu, const float* __restrict__ denom,
    const float* __restrict__ H, float* __restrict__ AGG, int E, int EN) {
    int e = blockIdx.x * 8 + (threadIdx.x >> 5);
    if (e >= EN) return;
    const int lane = threadIdx.x & 31;
    const int head = lane >> 3;
    int s, d; edge_sd(e, E, src, dst, s, d);
    float v = lrelu(asrc[s * HEADS + head] + adst[d * HEADS + head]);
    float m = fdec(maxu[d * HEADS + head]);
    float alpha = __expf(v - m) / (denom[d * HEADS + head] + 1e-16f);
    const float4 hv = ((const float4*)(H + (size_t)s * F))[lane];
    float* op = AGG + (size_t)d * F + lane * 4;
    atomicAdd(op + 0, alpha * hv.x);
    atomicAdd(op + 1, alpha * hv.y);
    atomicAdd(op + 2, alpha * hv.z);
    atomicAdd(op + 3, alpha * hv.w);
}

// ---- bias + relu -----------------------------------------------------------
__global__ __launch_bounds__(256) void bias_relu(
    const float* __restrict__ agg, const float* __restrict__ b,
    float* __restrict__ xout, int n) {
    int idx = blockIdx.x * blockDim.x + threadIdx.x;
    if (idx >= n * F) return;
    float v = agg[idx] + b[idx & (F - 1)];
    xout[idx] = v > 0.f ? v : 0.f;
}

// ---- global mean pool + FC -------------------------------------------------
__global__ __launch_bounds__(256) void zero_pool(float* __restrict__ pooled,
                                                 float* __restrict__ cnts) {
    int idx = blockIdx.x * blockDim.x + threadIdx.x;
    if (idx < NUM_GRAPHS * F) pooled[idx] = 0.f;
    if (idx < NUM_GRAPHS) cnts[idx] = 0.f;
}

__global__ __launch_bounds__(256) void pool_accum(
    const float* __restrict__ x, const int* __restrict__ batch,
    float* __restrict__ pooled, float* __restrict__ cnts, int n) {
    int idx = blockIdx.x * blockDim.x + threadIdx.x;
    if (idx >= n * F) return;
    const int node = idx >> 7, c = idx & (F - 1);
    const int g = batch[node];
    atomicAdd(&pooled[g * F + c], x[idx]);
    if (c == 0) atomicAdd(&cnts[g], 1.0f);
}

__global__ __launch_bounds__(256) void fc_kernel(
    const float* __restrict__ pooled, const float* __restrict__ cnts,
    const float* __restrict__ fcW, const float* __restrict__ fcb,
    float* __restrict__ out) {
    int idx = blockIdx.x * blockDim.x + threadIdx.x;
    if (idx >= NUM_GRAPHS * OUT_CH) return;
    const int g = idx / OUT_CH, o = idx % OUT_CH;
    float c = cnts[g]; c = c > 1.f ? c : 1.f;
    float s = 0.f;
#pragma unroll 4
    for (int k = 0; k < F; ++k) s += pooled[g * F + k] * fcW[k * OUT_CH + o];
    out[idx] = s / c + fcb[o];
}

// ---- one GAT layer ---------------------------------------------------------
static void run_gat_layer(hipStream_t stream, const float* xin, const float* W,
                          const float* a_s, const float* a_d, const float* b,
                          const int* srcp, const int* dstp, int N, int E,
                          float* bufH, float* xout, unsigned* maxu, float* denom,
                          float* asrc, float* adst, float* agg) {
    const int EN = E + N;
    gemm_xw_wmma<<<N / 16, 256, 0, stream>>>(xin, W, bufH);
    attn_coef<<<(N * HEADS + 255) / 256, 256, 0, stream>>>(bufH, a_s, a_d, asrc, adst, N);
    reset_layer<<<(N * F + 255) / 256, 256, 0, stream>>>(maxu, denom, agg, N);
    edge_max_pass<<<(EN + 255) / 256, 256, 0, stream>>>(srcp, dstp, asrc, adst, maxu, E, EN);
    edge_sum_pass<<<(EN + 255) / 256, 256, 0, stream>>>(srcp, dstp, asrc, adst, maxu, denom, E, EN);
    edge_agg_pass<<<(EN + 7) / 8, 256, 0, stream>>>(srcp, dstp, asrc, adst, maxu, denom,
                                                    bufH, agg, E, EN);
    bias_relu<<<(N * F + 255) / 256, 256, 0, stream>>>(agg, b, xout, N);
}

// ---- entry point -----------------------------------------------------------
extern "C" void kernel_launch(void* const* d_in, const int* in_sizes, int n_in,
                              void* d_out, int out_size, void* d_ws, size_t ws_size,
                              hipStream_t stream) {
    (void)n_in; (void)out_size; (void)ws_size;
    const float* x     = (const float*)d_in[0];
    const int*   ei    = (const int*)d_in[1];     // [2,E] flattened
    const int*   batch = (const int*)d_in[2];
    const float* W1  = (const float*)d_in[3];
    const float* a1s = (const float*)d_in[4];
    const float* a1d = (const float*)d_in[5];
    const float* b1  = (const float*)d_in[6];
    const float* W2  = (const float*)d_in[7];
    const float* a2s = (const float*)d_in[8];
    const float* a2d = (const float*)d_in[9];
    const float* b2  = (const float*)d_in[10];
    const float* W3  = (const float*)d_in[11];
    const float* a3s = (const float*)d_in[12];
    const float* a3d = (const float*)d_in[13];
    const float* b3  = (const float*)d_in[14];
    const float* fcW = (const float*)d_in[15];
    const float* fcb = (const float*)d_in[16];
    float* out = (float*)d_out;

    const int N = in_sizes[0] / F;    // 50000
    const int E = in_sizes[1] / 2;    // 800000
    const int* srcp = ei;
    const int* dstp = ei + E;

    // workspace carve-up
    char* ws = (char*)d_ws;
    size_t off = 0;
    auto carve = [&](size_t bytes) { void* p = ws + off; off += (bytes + 255) & ~(size_t)255; return p; };
    float*    bufX   = (float*)carve((size_t)N * F * sizeof(float));
    float*    bufH   = (float*)carve((size_t)N * F * sizeof(float));
    float*    bufAgg = (float*)carve((size_t)N * F * sizeof(float));
    float*    asrc   = (float*)carve((size_t)N * HEADS * sizeof(float));
    float*    adst   = (float*)carve((size_t)N * HEADS * sizeof(float));
    unsigned* maxu   = (unsigned*)carve((size_t)N * HEADS * sizeof(unsigned));
    float*    denom  = (float*)carve((size_t)N * HEADS * sizeof(float));
    float*    pooled = (float*)carve((size_t)NUM_GRAPHS * F * sizeof(float));
    float*    cnts   = (float*)carve((size_t)NUM_GRAPHS * sizeof(float));

    run_gat_layer(stream, x,    W1, a1s, a1d, b1, srcp, dstp, N, E,
                  bufH, bufX, maxu, denom, asrc, adst, bufAgg);
    run_gat_layer(stream, bufX, W2, a2s, a2d, b2, srcp, dstp, N, E,
                  bufH, bufX, maxu, denom, asrc, adst, bufAgg);
    run_gat_layer(stream, bufX, W3, a3s, a3d, b3, srcp, dstp, N, E,
                  bufH, bufX, maxu, denom, asrc, adst, bufAgg);

    zero_pool<<<(NUM_GRAPHS * F + 255) / 256, 256, 0, stream>>>(pooled, cnts);
    pool_accum<<<(N * F + 255) / 256, 256, 0, stream>>>(bufX, batch, pooled, cnts, N);
    fc_kernel<<<(NUM_GRAPHS * OUT_CH + 255) / 256, 256, 0, stream>>>(pooled, cnts, fcW, fcb, out);
}